// Router_1855425872526
// MI455X (gfx1250) — compile-verified
//
#include <hip/hip_runtime.h>

// MoE router: logits = hidden @ W^T  (M=16384, K=4096, N=64), softmax(E=64),
// top-2 + renormalize.  f32 WMMA (V_WMMA_F32_16X16X4_F32) keeps reference
// precision; the 256 MB f32 hidden read makes this memory-bound (~11 us floor
// at 23.3 TB/s), so no benefit from dropping to bf16.
//
// v2: hidden loads + all output stores are non-temporal (TH=NT) so the
// streamed-once 256 MB tensor doesn't evict the 1 MB gate matrix from
// WGP$/L2 (gate is re-read by all 1024 waves and must stay cache-resident);
// K-loop unrolled 8x for deeper load pipelining at low occupancy.

typedef __attribute__((ext_vector_type(2))) float v2f;
typedef __attribute__((ext_vector_type(4))) float v4f;
typedef __attribute__((ext_vector_type(8))) float v8f;

#define NB 4
#define NS 4096
#define NH 4096
#define NE 64
#define NTOK (NB * NS)          // 16384 tokens
#define WAVES_PER_BLOCK 8       // wave32 -> 256 threads
#define ROWPAD 68               // 64 cols + 4 pad floats (bank spread)

__global__ __launch_bounds__(256)
void router_gate_kernel(const float* __restrict__ hidden,
                        const float* __restrict__ gate,
                        float* __restrict__ out) {
    __shared__ float lds[WAVES_PER_BLOCK][16][ROWPAD];

    const int lane = threadIdx.x & 31;
    const int wave = threadIdx.x >> 5;
    const int tile = blockIdx.x * WAVES_PER_BLOCK + wave;  // 16-token tile
    const int row  = lane & 15;          // M row (A) / N col (B) within tile
    const int koff = (lane >> 4) << 1;   // lanes 0-15: K=0,1 ; lanes 16-31: K=2,3

    // A fragment (16x4 f32): one b64 load/lane.  B fragment (4x16 of W^T):
    // B[k][n] = W[n][k] -> same addressing pattern on the gate matrix.
    const float* aptr = hidden + (size_t)(tile * 16 + row) * NH + koff;
    const float* bp0  = gate + (size_t)( 0 + row) * NH + koff;
    const float* bp1  = gate + (size_t)(16 + row) * NH + koff;
    const float* bp2  = gate + (size_t)(32 + row) * NH + koff;
    const float* bp3  = gate + (size_t)(48 + row) * NH + koff;

    v8f acc0 = {}, acc1 = {}, acc2 = {}, acc3 = {};

#pragma unroll 8
    for (int k = 0; k < NH; k += 4) {
        // hidden is streamed exactly once -> non-temporal, keep gate cached
        v2f a  = __builtin_nontemporal_load((const v2f*)(aptr + k));
        v2f w0 = *(const v2f*)(bp0 + k);
        v2f w1 = *(const v2f*)(bp1 + k);
        v2f w2 = *(const v2f*)(bp2 + k);
        v2f w3 = *(const v2f*)(bp3 + k);
        acc0 = __builtin_amdgcn_wmma_f32_16x16x4_f32(false, a, false, w0,
                                                     (short)0, acc0, false, false);
        acc1 = __builtin_amdgcn_wmma_f32_16x16x4_f32(false, a, false, w1,
                                                     (short)0, acc1, false, false);
        acc2 = __builtin_amdgcn_wmma_f32_16x16x4_f32(false, a, false, w2,
                                                     (short)0, acc2, false, false);
        acc3 = __builtin_amdgcn_wmma_f32_16x16x4_f32(false, a, false, w3,
                                                     (short)0, acc3, false, false);
    }

    // Spill 16x64 logits to wave-private LDS slice.
    // C layout: VGPR r, lanes 0-15 -> M=r, N=lane ; lanes 16-31 -> M=8+r, N=lane-16
    const int mbase = (lane >> 4) << 3;
    const int col   = lane & 15;
#pragma unroll
    for (int r = 0; r < 8; ++r) {
        lds[wave][mbase + r][ 0 + col] = acc0[r];
        lds[wave][mbase + r][16 + col] = acc1[r];
        lds[wave][mbase + r][32 + col] = acc2[r];
        lds[wave][mbase + r][48 + col] = acc3[r];
    }
    // Wave-private LDS region: same-wave DS ops are ordered via DScnt waits
    // the compiler inserts; no cross-wave barrier needed.

    if (lane < 16) {
        const int m = lane;
        float* vals = &lds[wave][m][0];

        float mx = vals[0];
#pragma unroll
        for (int e = 1; e < NE; ++e) mx = fmaxf(mx, vals[e]);

        float sum = 0.f;
        float t1 = -1.f, t2 = -1.f;
        int   i1 = 0,   i2 = 0;
#pragma unroll
        for (int e = 0; e < NE; ++e) {
            float ex = __expf(vals[e] - mx);
            vals[e] = ex;
            sum += ex;
            if (ex > t1) { t2 = t1; i2 = i1; t1 = ex; i1 = e; }
            else if (ex > t2) { t2 = ex; i2 = e; }  // strict > keeps lowest idx on ties
        }

        const float inv   = 1.f / sum;
        const int   token = tile * 16 + m;
        float* probs = out + (size_t)token * NE;
#pragma unroll
        for (int e4 = 0; e4 < NE / 4; ++e4) {
            v4f p = *(const v4f*)&vals[e4 * 4];
            p *= inv;
            __builtin_nontemporal_store(p, (v4f*)(probs + e4 * 4));
        }

        // Outputs concatenated flat: probs | indices | weights
        float* idx_out = out + (size_t)NTOK * NE + (size_t)token * 2;
        float* w_out   = out + (size_t)NTOK * NE + (size_t)NTOK * 2 + (size_t)token * 2;
        const float rs = 1.f / (t1 + t2);   // renormalized top-2 weights
        v2f iv = { (float)i1, (float)i2 };
        v2f wv = { t1 * rs, t2 * rs };
        __builtin_nontemporal_store(iv, (v2f*)idx_out);
        __builtin_nontemporal_store(wv, (v2f*)w_out);
    }
}

extern "C" void kernel_launch(void* const* d_in, const int* in_sizes, int n_in,
                              void* d_out, int out_size, void* d_ws, size_t ws_size,
                              hipStream_t stream) {
    const float* hidden = (const float*)d_in[0];   // [B,S,H] f32
    const float* gate   = (const float*)d_in[1];   // [E,H]   f32
    float* out = (float*)d_out;

    const int tokens_per_block = 16 * WAVES_PER_BLOCK;          // 128
    const int grid = NTOK / tokens_per_block;                   // 128
    router_gate_kernel<<<grid, 256, 0, stream>>>(hidden, gate, out);
}